// RandomAffine_73959336837421
// MI455X (gfx1250) — compile-verified
//
#include <hip/hip_runtime.h>

// CDNA5 / gfx1250: wave32, WMMA via builtins.
typedef __attribute__((ext_vector_type(2))) float v2f;
typedef __attribute__((ext_vector_type(8))) float v8f;

#define IMG_H 2048
#define IMG_W 2048
#define TILES_X (IMG_W / 16)      // 128
#define WAVES_PER_BLOCK 8

// JAX 'reflect' index fixer, size = 2048 (power of two -> mask form).
__device__ __forceinline__ int reflect2k(int v) {
    int p = v & 4095;             // v mod 4096, correct for negatives (two's complement)
    return (p < 2048) ? p : (4095 - p);
}

__global__ __launch_bounds__(256)
void RandomAffine_affine_bilinear_kernel(const float* __restrict__ img,
                                         const float* __restrict__ M,
                                         float* __restrict__ out) {
    const int lane = threadIdx.x & 31;
    const int wave = threadIdx.x >> 5;
    const int tile = blockIdx.x * WAVES_PER_BLOCK + wave;   // 0..16383
    const int ti = tile >> 7;          // tile / 128
    const int tj = tile & 127;         // tile % 128
    const int i0 = ti << 4;
    const int j0 = tj << 4;

    // Uniform (scalar) loads of the 3x3 row-major affine matrix.
    const float m00 = M[0], m01 = M[1];
    const float m10 = M[3], m11 = M[4];
    const float m20 = M[6], m21 = M[7];

    const int half = lane >> 4;        // 0: rows 0..7 / K 0..1 ; 1: rows 8..15 / K 2..3
    const int col  = lane & 15;

    const float step = 2048.0f / 2047.0f;   // linspace(-1024, 1024, 2048) step

    // A (16x4 f32): row m = [y_m, 1, 0, 0].  Lanes 0-15 hold K=0,1; lanes 16-31 hold K=2,3 (zeros).
    float ym = -1024.0f + (float)(i0 + lane) * step;   // only meaningful for lane<16
    v2f A;
    A.x = (half == 0) ? ym   : 0.0f;
    A.y = (half == 0) ? 1.0f : 0.0f;

    // B (4x16 f32): column n = [c0, c1*x_n + c2, 0, 0].
    float xn = -1024.0f + (float)(j0 + col) * step;
    v2f By, Bx;
    By.x = (half == 0) ? m00 : 0.0f;
    By.y = (half == 0) ? (m10 * xn + m20 + 1024.0f) : 0.0f;
    Bx.x = (half == 0) ? m01 : 0.0f;
    Bx.y = (half == 0) ? (m11 * xn + m21 + 1024.0f) : 0.0f;

    v8f zero = {};
    // yy[m][n] = m00*y_m + m10*x_n + m20 + 1024 ; xx likewise.
    v8f yy = __builtin_amdgcn_wmma_f32_16x16x4_f32(false, A, false, By, (short)0, zero, false, false);
    v8f xx = __builtin_amdgcn_wmma_f32_16x16x4_f32(false, A, false, Bx, (short)0, zero, false, false);

    // D layout: VGPR r -> row (r + 8*half), column = lane&15.
    const int j = j0 + col;
#pragma unroll
    for (int r = 0; r < 8; ++r) {
        const int i = i0 + r + (half << 3);
        const float cy = yy[r];
        const float cx = xx[r];
        const float fy = floorf(cy);
        const float fx = floorf(cx);
        const float uy = cy - fy;      // upper weight (JAX: coordinate - floor)
        const float ux = cx - fx;
        const float ly = 1.0f - uy;    // lower weight
        const float lx = 1.0f - ux;
        const int iy = (int)fy;
        const int ix = (int)fx;
        const int y0 = reflect2k(iy);
        const int y1 = reflect2k(iy + 1);
        const int x0 = reflect2k(ix);
        const int x1 = reflect2k(ix + 1);

        const float* p00 = img + ((size_t)(y0 * IMG_W + x0)) * 3;
        const float* p01 = img + ((size_t)(y0 * IMG_W + x1)) * 3;
        const float* p10 = img + ((size_t)(y1 * IMG_W + x0)) * 3;
        const float* p11 = img + ((size_t)(y1 * IMG_W + x1)) * 3;
        float* o = out + ((size_t)(i * IMG_W + j)) * 3;

#pragma unroll
        for (int k = 0; k < 3; ++k) {
            const float v00 = p00[k];
            const float v01 = p01[k];
            const float v10 = p10[k];
            const float v11 = p11[k];
            o[k] = ly * (lx * v00 + ux * v01) + uy * (lx * v10 + ux * v11);
        }
    }
}

extern "C" void kernel_launch(void* const* d_in, const int* in_sizes, int n_in,
                              void* d_out, int out_size, void* d_ws, size_t ws_size,
                              hipStream_t stream) {
    (void)in_sizes; (void)n_in; (void)d_ws; (void)ws_size; (void)out_size;
    const float* img = (const float*)d_in[0];   // (2048, 2048, 3) f32
    const float* M   = (const float*)d_in[1];   // (3, 3) f32, row-major
    float* out = (float*)d_out;                 // (2048, 2048, 3) f32

    // 16384 tiles of 16x16; 8 waves (tiles) per 256-thread block -> 2048 blocks, no remainder.
    dim3 grid(2048), block(256);
    hipLaunchKernelGGL(RandomAffine_affine_bilinear_kernel, grid, block, 0, stream,
                       img, M, out);
}